// MultiTableImputer_47390669144622
// MI455X (gfx1250) — compile-verified
//
#include <hip/hip_runtime.h>

typedef __attribute__((ext_vector_type(2))) float v2f;
typedef __attribute__((ext_vector_type(8))) float v8f;

#define NA 200000
#define NB 50000
#define DIM 128

// ---------------------------------------------------------------------------
// native f32 L2 atomic add (no return) -> global_atomic_add_f32
// ---------------------------------------------------------------------------
__device__ __forceinline__ void gadd_f32(float* p, float v) {
    asm volatile("global_atomic_add_f32 %0, %1, off"
                 :
                 : "v"(p), "v"(v)
                 : "memory");
}

// ---------------------------------------------------------------------------
// zero workspace region (agg_a, agg_b, cnt_a, cnt_b contiguous)
// ---------------------------------------------------------------------------
__global__ __launch_bounds__(256) void zero_kernel(float4* __restrict__ p, long n4) {
    long i = (long)blockIdx.x * blockDim.x + threadIdx.x;
    long stride = (long)gridDim.x * blockDim.x;
    for (; i < n4; i += stride) p[i] = make_float4(0.f, 0.f, 0.f, 0.f);
}

// ---------------------------------------------------------------------------
// Y[rows,128] = X[rows,128] @ W[128,128]^T  (bias folded out; added at fuse)
// One wave per 16x16 output tile; 8 waves/block cover the 128 columns.
// A-matrix layout (32-bit, 16x4): lane L holds X[m=L%16][k0 + 2*(L>=16) + {0,1}]
// B-matrix symmetric with n = L%16.
// ---------------------------------------------------------------------------
__global__ __launch_bounds__(256) void transform_kernel(
    const float* __restrict__ X, const float* __restrict__ W,
    float* __restrict__ Y, int rows) {
    const int wave  = threadIdx.x >> 5;
    const int lane  = threadIdx.x & 31;
    const int rb    = blockIdx.x * 16;
    const int cb    = wave * 16;
    const int m     = lane & 15;          // also n for B
    const int khalf = (lane >> 4) * 2;    // 0 or 2
    if (rb >= rows) return;

    const float* xrow = X + (long)(rb + m) * DIM;
    const float* wrow = W + (long)(cb + m) * DIM;

    v8f c = {0.f, 0.f, 0.f, 0.f, 0.f, 0.f, 0.f, 0.f};
#pragma unroll
    for (int k0 = 0; k0 < DIM; k0 += 4) {
        v2f a = *(const v2f*)(xrow + k0 + khalf);
        v2f b = *(const v2f*)(wrow + k0 + khalf);
        c = __builtin_amdgcn_wmma_f32_16x16x4_f32(false, a, false, b,
                                                  (short)0, c, false, false);
    }
    // C/D layout: VGPR r -> M = r + 8*(lane>=16), N = lane%16
    const int mo = (lane >> 4) * 8;
#pragma unroll
    for (int r = 0; r < 8; ++r)
        Y[(long)(rb + r + mo) * DIM + cb + m] = c[r];
}

// ---------------------------------------------------------------------------
// Edge scatter: one wave per edge, lane handles 4 consecutive floats.
// fwd: agg_b[dst] += TA[src];  rev: agg_a[src] += TB[dst];  counts via lane 0.
// ---------------------------------------------------------------------------
__global__ __launch_bounds__(256) void edge_kernel(
    const int* __restrict__ src, const int* __restrict__ dst,
    const float* __restrict__ TA, const float* __restrict__ TB,
    float* __restrict__ agg_a, float* __restrict__ agg_b,
    float* __restrict__ cnt_a, float* __restrict__ cnt_b, int E) {
    const int wid  = (int)(((long)blockIdx.x * blockDim.x + threadIdx.x) >> 5);
    const int lane = threadIdx.x & 31;
    if (wid >= E) return;

    const int s = src[wid];
    const int d = dst[wid];
    const int fo = lane * 4;

    const float4 ta = *(const float4*)(TA + (long)s * DIM + fo);
    const float4 tb = *(const float4*)(TB + (long)d * DIM + fo);

    float* pb = agg_b + (long)d * DIM + fo;
    float* pa = agg_a + (long)s * DIM + fo;
    gadd_f32(pb + 0, ta.x); gadd_f32(pb + 1, ta.y);
    gadd_f32(pb + 2, ta.z); gadd_f32(pb + 3, ta.w);
    gadd_f32(pa + 0, tb.x); gadd_f32(pa + 1, tb.y);
    gadd_f32(pa + 2, tb.z); gadd_f32(pa + 3, tb.w);
    if (lane == 0) {
        gadd_f32(cnt_b + d, 1.0f);
        gadd_f32(cnt_a + s, 1.0f);
    }
}

// ---------------------------------------------------------------------------
// Fused gate + blend:
//   mean = (agg + cnt*b_msg) / max(cnt,1)
//   gate = sigmoid([row, mean] @ W_gate^T + b_gate)   (K=256 WMMA GEMM)
//   out  = gate*row + (1-gate)*mean
// ---------------------------------------------------------------------------
__global__ __launch_bounds__(256) void fuse_kernel(
    const float* __restrict__ row, const float* __restrict__ agg,
    const float* __restrict__ cnt, const float* __restrict__ Wg,
    const float* __restrict__ bg, const float* __restrict__ bmsg,
    float* __restrict__ out, int rows) {
    const int wave  = threadIdx.x >> 5;
    const int lane  = threadIdx.x & 31;
    const int rb    = blockIdx.x * 16;
    const int cb    = wave * 16;
    const int m     = lane & 15;
    const int khalf = (lane >> 4) * 2;
    if (rb >= rows) return;

    const float* xrow = row + (long)(rb + m) * DIM;
    const float* arow = agg + (long)(rb + m) * DIM;
    const float* wrow = Wg + (long)(cb + m) * (2 * DIM);  // W_gate row = 256 wide

    const float cn  = cnt[rb + m];
    const float inv = 1.0f / fmaxf(cn, 1.0f);

    v8f c = {0.f, 0.f, 0.f, 0.f, 0.f, 0.f, 0.f, 0.f};
    // K = 0..127 : row part of concat
#pragma unroll
    for (int k0 = 0; k0 < DIM; k0 += 4) {
        v2f a = *(const v2f*)(xrow + k0 + khalf);
        v2f b = *(const v2f*)(wrow + k0 + khalf);
        c = __builtin_amdgcn_wmma_f32_16x16x4_f32(false, a, false, b,
                                                  (short)0, c, false, false);
    }
    // K = 128..255 : mean part, built on the fly
#pragma unroll
    for (int k0 = 0; k0 < DIM; k0 += 4) {
        const int kk = k0 + khalf;
        v2f ag = *(const v2f*)(arow + kk);
        v2f bm = *(const v2f*)(bmsg + kk);
        v2f a;
        a.x = (ag.x + cn * bm.x) * inv;
        a.y = (ag.y + cn * bm.y) * inv;
        v2f b = *(const v2f*)(wrow + DIM + kk);
        c = __builtin_amdgcn_wmma_f32_16x16x4_f32(false, a, false, b,
                                                  (short)0, c, false, false);
    }

    // Epilogue: bias + sigmoid + blend. C layout: M = r + 8*(lane>=16), N = m.
    const int mo = (lane >> 4) * 8;
    const int f  = cb + m;
    const float bgf = bg[f];
    const float bmf = bmsg[f];
#pragma unroll
    for (int r = 0; r < 8; ++r) {
        const int g  = rb + r + mo;
        const float rv = row[(long)g * DIM + f];
        const float av = agg[(long)g * DIM + f];
        const float cv = cnt[g];
        const float mv = (av + cv * bmf) / fmaxf(cv, 1.0f);
        const float gate = 1.0f / (1.0f + __expf(-(c[r] + bgf)));
        out[(long)g * DIM + f] = gate * rv + (1.0f - gate) * mv;
    }
}

// ---------------------------------------------------------------------------
extern "C" void kernel_launch(void* const* d_in, const int* in_sizes, int n_in,
                              void* d_out, int out_size, void* d_ws, size_t ws_size,
                              hipStream_t stream) {
    const float* A      = (const float*)d_in[0];
    const float* B      = (const float*)d_in[1];
    const float* W_fwd  = (const float*)d_in[2];
    const float* b_fwd  = (const float*)d_in[3];
    const float* W_rev  = (const float*)d_in[4];
    const float* b_rev  = (const float*)d_in[5];
    const float* W_gate = (const float*)d_in[6];
    const float* b_gate = (const float*)d_in[7];
    const int*   src    = (const int*)d_in[8];
    const int*   dst    = (const int*)d_in[9];
    const int    E      = in_sizes[8];

    // workspace layout (floats): TA | TB | agg_a | agg_b | cnt_a | cnt_b
    float* ws    = (float*)d_ws;
    float* TA    = ws;
    float* TB    = TA + (size_t)NA * DIM;
    float* agg_a = TB + (size_t)NB * DIM;
    float* agg_b = agg_a + (size_t)NA * DIM;
    float* cnt_a = agg_b + (size_t)NB * DIM;
    float* cnt_b = cnt_a + NA;

    // zero agg_a..cnt_b (contiguous)
    const long zf = (long)NA * DIM + (long)NB * DIM + NA + NB;  // 32,250,000
    zero_kernel<<<2048, 256, 0, stream>>>((float4*)agg_a, zf / 4);

    // per-row linear transforms (bias deferred)
    transform_kernel<<<NA / 16, 256, 0, stream>>>(A, W_fwd, TA, NA);
    transform_kernel<<<NB / 16, 256, 0, stream>>>(B, W_rev, TB, NB);

    // edge gather + scatter-add (one wave per edge)
    const int nthreads_edges = E * 32;
    edge_kernel<<<(nthreads_edges + 255) / 256, 256, 0, stream>>>(
        src, dst, TA, TB, agg_a, agg_b, cnt_a, cnt_b, E);

    // fused mean/gate/blend per table (A receives rev msgs -> b_rev; B -> b_fwd)
    float* out_a = (float*)d_out;
    float* out_b = out_a + (size_t)NA * DIM;
    fuse_kernel<<<NA / 16, 256, 0, stream>>>(A, agg_a, cnt_a, W_gate, b_gate, b_rev, out_a, NA);
    fuse_kernel<<<NB / 16, 256, 0, stream>>>(B, agg_b, cnt_b, W_gate, b_gate, b_fwd, out_b, NB);
}